// Head_5798205850017
// MI455X (gfx1250) — compile-verified
//
#include <hip/hip_runtime.h>
#include <hip/hip_bf16.h>

// ---------------------------------------------------------------------------
// MI455X / gfx1250 implementation.
// fp32 end-to-end with V_WMMA_F32_16X16X4_F32 for all GEMM-shaped work.
// Branch-free inner loops: out-of-range rows/cols are handled by clamping
// load addresses (garbage only reaches store-guarded rows/cols of D).
// Algebraic fusion: (softmax(qk^T) v) @ ep_w == softmax(qk^T) (v @ ep_w).
// ---------------------------------------------------------------------------

typedef float v2f __attribute__((ext_vector_type(2)));
typedef float v8f __attribute__((ext_vector_type(8)));

#define B_ 8
#define D_ 49
#define C_ 512
#define H_ 256     // C/2
#define N_ 12
#define E_ 144     // N*N
#define EPS_ 1e-5f
#define SCALE_ 0.0625f   // (C/2)^-0.5

__device__ __forceinline__ v8f wmma4(v2f a, v2f b, v8f c) {
  return __builtin_amdgcn_wmma_f32_16x16x4_f32(false, a, false, b, (short)0, c,
                                               false, false);
}

__device__ __forceinline__ float wred(float v) {
#pragma unroll
  for (int o = 16; o > 0; o >>= 1) v += __shfl_xor(v, o, 32);
  return v;
}
__device__ __forceinline__ float wmaxr(float v) {
#pragma unroll
  for (int o = 16; o > 0; o >>= 1) v = fmaxf(v, __shfl_xor(v, o, 32));
  return v;
}

__device__ __forceinline__ void block_red2(float& s, float& s2, float* red, int tid) {
  int lane = tid & 31, wid = tid >> 5;
  s = wred(s); s2 = wred(s2);
  if (lane == 0) { red[wid] = s; red[8 + wid] = s2; }
  __syncthreads();
  if (tid == 0) {
    float a = 0.f, b = 0.f;
    for (int w = 0; w < 8; ++w) { a += red[w]; b += red[8 + w]; }
    red[0] = a; red[8] = b;
  }
  __syncthreads();
  s = red[0]; s2 = red[8];
}

// ---------------------------------------------------------------------------
// Generic batched GEMM: C = A@B (+bias).  One wave per 16x32 output block
// (two 16x16 WMMA tiles sharing the A fragment).  K % 4 == 0.
// Loads are branch-free: addresses clamped, edge rows/cols guarded at store.
// ---------------------------------------------------------------------------
__global__ void __launch_bounds__(256)
gemm_wmma(const float* __restrict__ A, const float* __restrict__ Bm,
          const float* __restrict__ bias, float* __restrict__ Cc,
          int M, int Nn, int K, int lda, int ldb, int ldc,
          long sA, long sB, long sC, long sBias) {
  int batch = blockIdx.y;
  A += (long)batch * sA;
  Bm += (long)batch * sB;
  Cc += (long)batch * sC;
  if (bias) bias += (long)batch * sBias;

  int tilesM = (M + 15) >> 4;
  int tilesN = (Nn + 15) >> 4;
  int tilesN2 = (tilesN + 1) >> 1;
  int wave = blockIdx.x * 8 + (threadIdx.x >> 5);
  int lane = threadIdx.x & 31;
  if (wave >= tilesM * tilesN2) return;
  int tm = wave / tilesN2, tn2 = wave % tilesN2;
  int tn0 = tn2 * 2, tn1 = tn0 + 1;
  bool has2 = tn1 < tilesN;

  int l15 = lane & 15;
  int ko  = (lane >> 4) << 1;                  // K sub-offset per half-wave
  int row_a = tm * 16 + l15;
  int row_cl = row_a < M ? row_a : M - 1;      // clamp: garbage rows guarded at store
  int col0 = tn0 * 16 + l15;
  int col1 = tn1 * 16 + l15;
  int col0_cl = col0 < Nn ? col0 : Nn - 1;
  int col1_cl = (has2 && col1 < Nn) ? col1 : col0_cl;

  const float* Ap  = A + (long)row_cl * lda + ko;
  const float* Bp0 = Bm + col0_cl + (long)ko * ldb;
  const float* Bp1 = Bm + col1_cl + (long)ko * ldb;

  v8f acc0 = {0.f, 0.f, 0.f, 0.f, 0.f, 0.f, 0.f, 0.f};
  v8f acc1 = {0.f, 0.f, 0.f, 0.f, 0.f, 0.f, 0.f, 0.f};
#pragma unroll 2
  for (int k0 = 0; k0 < K; k0 += 4) {
    v2f a = *(const v2f*)(Ap + k0);
    v2f b0, b1;
    b0.x = Bp0[(long)k0 * ldb];
    b0.y = Bp0[(long)k0 * ldb + ldb];
    b1.x = Bp1[(long)k0 * ldb];
    b1.y = Bp1[(long)k0 * ldb + ldb];
    acc0 = wmma4(a, b0, acc0);
    acc1 = wmma4(a, b1, acc1);
  }

  int rb = tm * 16 + ((lane >> 4) << 3);
  bool fullM = (tm * 16 + 16) <= M;
  if (col0 < Nn) {
    float bv = bias ? bias[col0] : 0.f;
    if (fullM) {
#pragma unroll
      for (int r = 0; r < 8; ++r) Cc[(long)(rb + r) * ldc + col0] = acc0[r] + bv;
    } else {
#pragma unroll
      for (int r = 0; r < 8; ++r)
        if (rb + r < M) Cc[(long)(rb + r) * ldc + col0] = acc0[r] + bv;
    }
  }
  if (has2 && col1 < Nn) {
    float bv = bias ? bias[col1] : 0.f;
    if (fullM) {
#pragma unroll
      for (int r = 0; r < 8; ++r) Cc[(long)(rb + r) * ldc + col1] = acc1[r] + bv;
    } else {
#pragma unroll
      for (int r = 0; r < 8; ++r)
        if (rb + r < M) Cc[(long)(rb + r) * ldc + col1] = acc1[r] + bv;
    }
  }
}

// ---------------------------------------------------------------------------
// Attention helper: S(64x64 LDS, zero padded) = softmax_rows(q k^T * SCALE),
// valid region 49x49.  q,k are (49,256) row-major; the B operand is k^T, so
// both fragments are contiguous b64 loads.
// ---------------------------------------------------------------------------
__device__ __forceinline__ void attn_scores_softmax(const float* q, const float* k,
                                                    float* S, int tid) {
  int lane = tid & 31, wid = tid >> 5;
  for (int idx = tid; idx < 64 * 64; idx += 256) S[idx] = 0.f;
  __syncthreads();

  int l15 = lane & 15;
  int ko = (lane >> 4) << 1;
  for (int t = wid; t < 16; t += 8) {              // 4x4 tiles of 16x16
    int tm = t >> 2, tn = t & 3;
    int row = tm * 16 + l15;
    int col = tn * 16 + l15;
    int rcl = row < D_ ? row : D_ - 1;
    int ccl = col < D_ ? col : D_ - 1;
    const float* qp = q + rcl * H_ + ko;
    const float* kp = k + ccl * H_ + ko;
    v8f acc = {0.f, 0.f, 0.f, 0.f, 0.f, 0.f, 0.f, 0.f};
#pragma unroll 4
    for (int k0 = 0; k0 < H_; k0 += 4) {
      v2f a = *(const v2f*)(qp + k0);
      v2f b = *(const v2f*)(kp + k0);
      acc = wmma4(a, b, acc);
    }
    if (col < D_) {
      int rb = tm * 16 + ((lane >> 4) << 3);
      if (tm < 3) {
#pragma unroll
        for (int r = 0; r < 8; ++r) S[(rb + r) * 64 + col] = acc[r] * SCALE_;
      } else {
#pragma unroll
        for (int r = 0; r < 8; ++r)
          if (rb + r < D_) S[(rb + r) * 64 + col] = acc[r] * SCALE_;
      }
    }
  }
  __syncthreads();

  for (int row = wid; row < D_; row += 8) {        // softmax over 49 cols
    float v0 = (lane < D_)      ? S[row * 64 + lane]      : -1e30f;
    float v1 = (lane + 32 < D_) ? S[row * 64 + lane + 32] : -1e30f;
    float m = wmaxr(fmaxf(v0, v1));
    float e0 = (lane < D_)      ? __expf(v0 - m) : 0.f;
    float e1 = (lane + 32 < D_) ? __expf(v1 - m) : 0.f;
    float inv = 1.f / wred(e0 + e1);
    if (lane < D_)      S[row * 64 + lane]      = e0 * inv;
    if (lane + 32 < D_) S[row * 64 + lane + 32] = e1 * inv;
  }
  __syncthreads();
}

// ------------------- fam cross attention: feat1 = S @ v --------------------
// grid.x = N_*B_ (block index = n*B_+b)
__global__ void __launch_bounds__(256)
fam_attn_kernel(const float* __restrict__ qf, const float* __restrict__ kf,
                const float* __restrict__ vf, float* __restrict__ feat1) {
  __shared__ float S[64 * 64];
  int n = blockIdx.x / B_, b = blockIdx.x % B_;
  const float* q = qf + (long)(n * B_ + b) * D_ * H_;
  const float* k = kf + (long)b * D_ * H_;
  const float* v = vf + (long)b * D_ * C_;
  float* out = feat1 + (long)(n * B_ + b) * D_ * C_;
  int tid = threadIdx.x, lane = tid & 31, wid = tid >> 5;
  int l15 = lane & 15;
  int ko = (lane >> 4) << 1;

  attn_scores_softmax(q, k, S, tid);

  for (int t = wid; t < 4 * 32; t += 8) {          // 64x512 output, K=64 padded
    int tm = t >> 5, tn = t & 31;
    int row = tm * 16 + l15;
    int col = tn * 16 + l15;
    const float* sp = S + row * 64 + ko;           // row < 64: always valid
    const float* vp = v + col;
    v8f acc = {0.f, 0.f, 0.f, 0.f, 0.f, 0.f, 0.f, 0.f};
#pragma unroll 2
    for (int k0 = 0; k0 < 64; k0 += 4) {
      int ka = k0 + ko;
      int k1 = ka < D_ ? ka : D_ - 1;              // S col >= 49 is 0: 0*x = 0
      int k2 = (ka + 1) < D_ ? (ka + 1) : D_ - 1;
      v2f a = *(const v2f*)(sp + k0);
      v2f bb;
      bb.x = vp[k1 * C_];
      bb.y = vp[k2 * C_];
      acc = wmma4(a, bb, acc);
    }
    int rb = tm * 16 + ((lane >> 4) << 3);
    if (tm < 3) {
#pragma unroll
      for (int r = 0; r < 8; ++r) out[(rb + r) * C_ + col] = acc[r];
    } else {
#pragma unroll
      for (int r = 0; r < 8; ++r)
        if (rb + r < D_) out[(rb + r) * C_ + col] = acc[r];
    }
  }
}

// ---- arm attention fused with ep projection (via vW = v@ep_w precompute) ---
// grid.x = B_*E_ (block = b*E_+e).  Accumulates fe_sum[b,e,c] = sum_d z and
// per-edge BN sums of z, z^2 where z = (softmax(qk^T) vW)[d,c].
__global__ void __launch_bounds__(256)
arm_attn_kernel(const float* __restrict__ qa, const float* __restrict__ ka,
                const float* __restrict__ vw, float* __restrict__ fe_sum,
                float* __restrict__ est1, float* __restrict__ est2) {
  __shared__ float S[64 * 64];
  int b = blockIdx.x / E_, e = blockIdx.x % E_;
  int i = e / N_, j = e % N_;                       // end=i (k,v), start=j (q)
  const float* q = qa + (long)(j * B_ + b) * D_ * H_;
  const float* k = ka + (long)(i * B_ + b) * D_ * H_;
  const float* v = vw + (long)(i * B_ + b) * D_ * C_;
  float* fes = fe_sum + (long)(b * E_ + e) * C_;
  int tid = threadIdx.x, lane = tid & 31, wid = tid >> 5;
  int l15 = lane & 15;
  int ko = (lane >> 4) << 1;

  attn_scores_softmax(q, k, S, tid);

  float zs = 0.f, zq = 0.f;
  for (int t = wid; t < 4 * 32; t += 8) {
    int tm = t >> 5, tn = t & 31;
    int row = tm * 16 + l15;
    int col = tn * 16 + l15;
    const float* sp = S + row * 64 + ko;
    const float* vp = v + col;
    v8f acc = {0.f, 0.f, 0.f, 0.f, 0.f, 0.f, 0.f, 0.f};
#pragma unroll 2
    for (int k0 = 0; k0 < 64; k0 += 4) {
      int ka2 = k0 + ko;
      int k1 = ka2 < D_ ? ka2 : D_ - 1;
      int k2 = (ka2 + 1) < D_ ? (ka2 + 1) : D_ - 1;
      v2f a = *(const v2f*)(sp + k0);
      v2f bb;
      bb.x = vp[k1 * C_];
      bb.y = vp[k2 * C_];
      acc = wmma4(a, bb, acc);
    }
    float colsum = 0.f;
    int rb = tm * 16 + ((lane >> 4) << 3);
#pragma unroll
    for (int r = 0; r < 8; ++r) {
      int rr = rb + r;
      if (rr < D_) { float z = acc[r]; colsum += z; zs += z; zq += z * z; }
    }
    colsum += __shfl_xor(colsum, 16, 32);          // lanes l, l^16 share a col
    if (lane < 16) atomicAdd(&fes[tn * 16 + lane], colsum);
  }
  zs = wred(zs); zq = wred(zq);
  if (lane == 0) { atomicAdd(&est1[e], zs); atomicAdd(&est2[e], zq); }
}

// ---------------- BN stats for h over (b,d) per (n,c) ----------------------
// grid = N_ ; thread-per-column, rows looped (fully coalesced).
__global__ void __launch_bounds__(256)
bn1_stats_kernel(const float* __restrict__ h, float* __restrict__ s1,
                 float* __restrict__ s2) {
  int n = blockIdx.x;
  const float* p = h + (long)n * (B_ * D_) * C_;
  for (int c = threadIdx.x; c < C_; c += 256) {
    float s = 0.f, sq = 0.f;
    for (int r = 0; r < B_ * D_; ++r) {
      float t = p[(long)r * C_ + c];
      s += t; sq += t * t;
    }
    s1[n * C_ + c] = s;
    s2[n * C_ + c] = sq;
  }
}

// ------------ normalize+relu h (in place -> f_u) and f_v = mean_d ----------
// grid = N_*B_*(C_/256)
__global__ void __launch_bounds__(256)
bn1_apply_kernel(float* __restrict__ h, const float* __restrict__ s1,
                 const float* __restrict__ s2, float* __restrict__ f_v) {
  int blk = blockIdx.x;
  int cblk = blk & 1;
  int nb = blk >> 1;
  int b = nb % B_, n = nb / B_;
  int c = cblk * 256 + threadIdx.x;
  float cnt = (float)(B_ * D_);
  float m = s1[n * C_ + c] / cnt;
  float var = s2[n * C_ + c] / cnt - m * m;
  float rstd = rsqrtf(var + EPS_);
  float* base = h + ((long)(n * B_ + b)) * D_ * C_ + c;
  float acc = 0.f;
  for (int d = 0; d < D_; ++d) {
    float t = fmaxf((base[(long)d * C_] - m) * rstd, 0.f);
    base[(long)d * C_] = t;
    acc += t;
  }
  f_v[(long)(b * N_ + n) * C_ + c] = acc / (float)D_;
}

// ----------------------- cosine-similarity mask ----------------------------
__global__ void __launch_bounds__(256)
mask_kernel(const float* __restrict__ f_v, float* __restrict__ mask) {
  __shared__ float si[N_ * C_];
  __shared__ float adj[N_ * N_];
  __shared__ float dis[N_];
  int b = blockIdx.x;
  int tid = threadIdx.x, lane = tid & 31, wid = tid >> 5;
  for (int i = wid; i < N_; i += 8) {
    const float* fv = f_v + (long)(b * N_ + i) * C_;
    float s = 0.f;
    for (int c = lane; c < C_; c += 32) { float t = fv[c]; s += t * t; }
    s = wred(s);
    float inv = 1.f / fmaxf(sqrtf(s), 1e-12f);
    for (int c = lane; c < C_; c += 32) si[i * C_ + c] = fv[c] * inv;
  }
  __syncthreads();
  for (int p = wid; p < N_ * N_; p += 8) {
    int i = p / N_, jj = p % N_;
    float s = 0.f;
    for (int c = lane; c < C_; c += 32) s += si[i * C_ + c] * si[jj * C_ + c];
    s = wred(s);
    if (lane == 0) adj[p] = s;
  }
  __syncthreads();
  if (tid < N_) {
    float s = 0.f;
    for (int jj = 0; jj < N_; ++jj) s += adj[tid * N_ + jj];
    dis[tid] = rsqrtf(s);
  }
  __syncthreads();
  for (int p = tid; p < N_ * N_; p += 256)
    mask[b * E_ + p] = adj[p] * dis[p / N_] * dis[p % N_];
}

// ------- finalize f_e: add ep_b analytically, per-edge BN, mean_d, mask ----
__global__ void __launch_bounds__(256)
fe_finalize_kernel(const float* __restrict__ fe_sum, const float* __restrict__ est1,
                   const float* __restrict__ est2, const float* __restrict__ ep_b,
                   const float* __restrict__ mask, float* __restrict__ f_e) {
  __shared__ float red[24];
  int e = blockIdx.x;
  int tid = threadIdx.x, lane = tid & 31, wid = tid >> 5;
  float dot = 0.f, se = 0.f, se2 = 0.f;
  for (int c = tid; c < C_; c += 256) {
    float zc = 0.f;
    for (int b = 0; b < B_; ++b) zc += fe_sum[(long)(b * E_ + e) * C_ + c];
    float eb = ep_b[c];
    dot += eb * zc; se += eb; se2 += eb * eb;
  }
  dot = wred(dot); se = wred(se); se2 = wred(se2);
  if (lane == 0) { red[wid] = dot; red[8 + wid] = se; red[16 + wid] = se2; }
  __syncthreads();
  if (tid == 0) {
    float a = 0.f, b2 = 0.f, c3 = 0.f;
    for (int w = 0; w < 8; ++w) { a += red[w]; b2 += red[8 + w]; c3 += red[16 + w]; }
    red[0] = a; red[8] = b2; red[16] = c3;
  }
  __syncthreads();
  dot = red[0]; se = red[8]; se2 = red[16];
  float cnt = (float)(B_ * D_ * C_);
  float bd = (float)(B_ * D_);
  float Sy = est1[e] + bd * se;                    // y = z + ep_b
  float Sy2 = est2[e] + 2.f * dot + bd * se2;
  float m = Sy / cnt;
  float var = Sy2 / cnt - m * m;
  float rstd = rsqrtf(var + EPS_);
  for (int c = tid; c < C_; c += 256) {
    float eb = ep_b[c];
    for (int b = 0; b < B_; ++b) {
      float z = fe_sum[(long)(b * E_ + e) * C_ + c] / (float)D_ + eb;
      f_e[(long)(b * E_ + e) * C_ + c] = (z - m) * rstd * mask[b * E_ + e];
    }
  }
}

// -------------------- GNN: edge update (per edge e) ------------------------
__global__ void __launch_bounds__(256)
gnn_edge_kernel(const float* __restrict__ ta, const float* __restrict__ tb,
                const float* __restrict__ te, float* __restrict__ ed) {
  __shared__ float aggs[B_ * C_];                  // 16 KB
  __shared__ float red[16];
  int e = blockIdx.x;
  int i = e / N_, j = e % N_;
  int tid = threadIdx.x;
  float s = 0.f, s2 = 0.f;
  for (int idx = tid; idx < B_ * C_; idx += 256) {
    int b = idx >> 9, c = idx & (C_ - 1);
    float v = ta[(long)(b * N_ + i) * C_ + c] + tb[(long)(b * N_ + j) * C_ + c] +
              te[(long)(b * E_ + e) * C_ + c];
    aggs[idx] = v;
    s += v; s2 += v * v;
  }
  block_red2(s, s2, red, tid);
  float cnt = (float)(B_ * C_);
  float m = s / cnt;
  float rstd = rsqrtf(s2 / cnt - m * m + EPS_);
  for (int idx = tid; idx < B_ * C_; idx += 256) {
    int b = idx >> 9, c = idx & (C_ - 1);
    long o = (long)(b * E_ + e) * C_ + c;
    ed[o] += fmaxf((aggs[idx] - m) * rstd, 0.f);
  }
}

// -------- GNN: gated message  msg[b,i,c] = sum_j softmax_j(sig(ed)) * tv / N
__global__ void __launch_bounds__(256)
gnn_msg_kernel(const float* __restrict__ ed, const float* __restrict__ tv,
               float* __restrict__ msg) {
  int cblk = blockIdx.x & 1;
  int idx = blockIdx.x >> 1;
  int i = idx % N_, b = idx / N_;
  int c = cblk * 256 + threadIdx.x;
  float wsum = 0.f, acc = 0.f;
  for (int j = 0; j < N_; ++j) {
    float sg = 1.f / (1.f + __expf(-ed[(long)(b * E_ + i * N_ + j) * C_ + c]));
    float w = __expf(sg);                          // sigmoid in (0,1): safe
    wsum += w;
    acc += w * tv[(long)(b * N_ + j) * C_ + c];
  }
  msg[(long)(b * N_ + i) * C_ + c] = acc / (wsum * (float)N_);
}

// -------------------- GNN: node update (per node nn) -----------------------
__global__ void __launch_bounds__(256)
gnn_node_kernel(const float* __restrict__ tu, const float* __restrict__ msg,
                float* __restrict__ xv) {
  __shared__ float us[B_ * C_];
  __shared__ float red[16];
  int nn = blockIdx.x;
  int tid = threadIdx.x;
  float s = 0.f, s2 = 0.f;
  for (int idx = tid; idx < B_ * C_; idx += 256) {
    int b = idx >> 9, c = idx & (C_ - 1);
    long o = (long)(b * N_ + nn) * C_ + c;
    float v = tu[o] + msg[o];
    us[idx] = v;
    s += v; s2 += v * v;
  }
  block_red2(s, s2, red, tid);
  float cnt = (float)(B_ * C_);
  float m = s / cnt;
  float rstd = rsqrtf(s2 / cnt - m * m + EPS_);
  for (int idx = tid; idx < B_ * C_; idx += 256) {
    int b = idx >> 9, c = idx & (C_ - 1);
    long o = (long)(b * N_ + nn) * C_ + c;
    xv[o] = fmaxf(xv[o] + (us[idx] - m) * rstd, 0.f);
  }
}

// ----------------------- cosine classifier ---------------------------------
__global__ void __launch_bounds__(256)
classify_kernel(const float* __restrict__ xv, const float* __restrict__ sc,
                float* __restrict__ out) {
  int w = blockIdx.x * 8 + (threadIdx.x >> 5);
  int lane = threadIdx.x & 31;
  if (w >= B_ * N_) return;
  int b = w / N_, n = w % N_;
  const float* f = xv + (long)(b * N_ + n) * C_;
  const float* s = sc + (long)n * C_;
  float nf = 0.f, ns = 0.f, dp = 0.f;
  for (int c = lane; c < C_; c += 32) {
    float fv = f[c];
    float sv = fmaxf(s[c], 0.f);
    nf += fv * fv; ns += sv * sv; dp += fv * sv;
  }
  nf = wred(nf); ns = wred(ns); dp = wred(dp);
  if (lane == 0)
    out[b * N_ + n] = dp / (fmaxf(sqrtf(nf), 1e-12f) * fmaxf(sqrtf(ns), 1e-12f));
}

// ===========================================================================
extern "C" void kernel_launch(void* const* d_in, const int* in_sizes, int n_in,
                              void* d_out, int out_size, void* d_ws, size_t ws_size,
                              hipStream_t stream) {
  (void)in_sizes; (void)n_in; (void)out_size; (void)ws_size;
  const float* x      = (const float*)d_in[0];
  const float* Wc     = (const float*)d_in[1];
  const float* bc     = (const float*)d_in[2];
  const float* fam_wq = (const float*)d_in[3];
  const float* fam_bq = (const float*)d_in[4];
  const float* fam_wk = (const float*)d_in[5];
  const float* fam_bk = (const float*)d_in[6];
  const float* fam_wv = (const float*)d_in[7];
  const float* fam_bv = (const float*)d_in[8];
  const float* arm_wq = (const float*)d_in[9];
  const float* arm_bq = (const float*)d_in[10];
  const float* arm_wk = (const float*)d_in[11];
  const float* arm_bk = (const float*)d_in[12];
  const float* arm_wv = (const float*)d_in[13];
  const float* arm_bv = (const float*)d_in[14];
  const float* ep_w   = (const float*)d_in[15];
  const float* ep_b   = (const float*)d_in[16];
  const float* g_w[10];
  for (int k = 0; k < 10; ++k) g_w[k] = (const float*)d_in[17 + k];
  const float* sc     = (const float*)d_in[27];
  const float* efc_w  = (const float*)d_in[28];
  const float* efc_b  = (const float*)d_in[29];
  float* out = (float*)d_out;

  // -------- workspace layout (floats) --------
  float* ws = (float*)d_ws;
  float* f_h    = ws + 0;         // (N,B,D,C) h -> f_u       2,408,448
  float* q_arm  = ws + 0;         // reuses f_h after fam stage
  float* k_arm  = ws + 1204224;
  float* q_fam  = ws + 2408448;   // (N,B,D,H)                 1,204,224
  float* k_fam  = ws + 3612672;   // (B,D,H)                     100,352
  float* v_fam  = ws + 3713024;   // (B,D,C)                     200,704
  float* feat1  = ws + 3913728;   // (N,B,D,C)                 2,408,448
  float* v_arm  = ws + 6322176;   // (N,B,D,C)                 2,408,448
  float* vW     = ws + 8730624;   // v_arm @ ep_w              2,408,448
  float* fe_sum = ws + 11139072;  // (B,E,C)                     589,824
  float* f_e    = ws + 11728896;  // (B,E,C)                     589,824
  float* f_v    = ws + 12318720;  // (B,N,C)                      49,152
  float* t_a    = ws + 12367872;
  float* t_b    = ws + 12417024;
  float* t_u    = ws + 12466176;
  float* t_v    = ws + 12515328;
  float* msg    = ws + 12564480;
  float* t_e    = ws + 12613632;  // (B,E,C)                     589,824
  float* est1   = ws + 13203456;  // (E)
  float* est2   = ws + 13203600;  // (E)
  float* maskb  = ws + 13203744;  // (B,E)
  float* s1     = ws + 13204896;  // (N,C)
  float* s2     = ws + 13211040;  // (N,C)  end ~ 52.9 MB

  auto gemm = [&](const float* A, int lda, const float* Bm, int ldb,
                  const float* bias, float* Cc, int ldc, int M, int Nn, int K,
                  int batch, long sA, long sB, long sBias, long sC) {
    int tilesN2 = (((Nn + 15) / 16) + 1) / 2;
    int tiles = ((M + 15) / 16) * tilesN2;
    dim3 grid((tiles + 7) / 8, batch);
    gemm_wmma<<<grid, 256, 0, stream>>>(A, Bm, bias, Cc, M, Nn, K, lda, ldb,
                                        ldc, sA, sB, sC, sBias);
  };

  // 1) per-class h = x @ Wc[n] + bc[n]   (batched over N)
  gemm(x, C_, Wc, C_, bc, f_h, C_, B_ * D_, C_, C_,
       N_, 0, (long)C_ * C_, C_, (long)B_ * D_ * C_);
  // 2) BN over (b,d) per (n,c); relu -> f_u (in place); f_v = mean_d
  bn1_stats_kernel<<<N_, 256, 0, stream>>>(f_h, s1, s2);
  bn1_apply_kernel<<<N_ * B_ * (C_ / 256), 256, 0, stream>>>(f_h, s1, s2, f_v);
  // 3) fam projections (k,v shared over N since xkv is broadcast x)
  gemm(x, C_, fam_wk, H_, fam_bk, k_fam, H_, B_ * D_, H_, C_, 1, 0, 0, 0, 0);
  gemm(x, C_, fam_wv, C_, fam_bv, v_fam, C_, B_ * D_, C_, C_, 1, 0, 0, 0, 0);
  gemm(f_h, C_, fam_wq, H_, fam_bq, q_fam, H_, N_ * B_ * D_, H_, C_, 1, 0, 0, 0, 0);
  // 4) fam cross-attention -> feat1
  fam_attn_kernel<<<N_ * B_, 256, 0, stream>>>(q_fam, k_fam, v_fam, feat1);
  // 5) arm projections (only N distinct since start=tile, end=repeat)
  gemm(feat1, C_, arm_wq, H_, arm_bq, q_arm, H_, N_ * B_ * D_, H_, C_, 1, 0, 0, 0, 0);
  gemm(feat1, C_, arm_wk, H_, arm_bk, k_arm, H_, N_ * B_ * D_, H_, C_, 1, 0, 0, 0, 0);
  gemm(feat1, C_, arm_wv, C_, arm_bv, v_arm, C_, N_ * B_ * D_, C_, C_, 1, 0, 0, 0, 0);
  // 6) vW = v_arm @ ep_w  (associativity fusion; ep_b folded analytically)
  gemm(v_arm, C_, ep_w, C_, nullptr, vW, C_, N_ * B_ * D_, C_, C_, 1, 0, 0, 0, 0);
  // 7) per-edge arm attention + ep accumulation
  hipMemsetAsync(fe_sum, 0, (size_t)B_ * E_ * C_ * sizeof(float), stream);
  hipMemsetAsync(est1, 0, (size_t)2 * E_ * sizeof(float), stream);
  arm_attn_kernel<<<B_ * E_, 256, 0, stream>>>(q_arm, k_arm, vW, fe_sum, est1, est2);
  // 8) cosine adjacency mask, then f_e finalize (BN + mean_d + mask)
  mask_kernel<<<B_, 256, 0, stream>>>(f_v, maskb);
  fe_finalize_kernel<<<E_, 256, 0, stream>>>(fe_sum, est1, est2, ep_b, maskb, f_e);
  // 9) two GNN message-passing layers (in place on f_v / f_e)
  auto gnn_layer = [&](const float* wa, const float* wb, const float* we,
                       const float* wu, const float* wv) {
    gemm(f_v, C_, wa, C_, nullptr, t_a, C_, B_ * N_, C_, C_, 1, 0, 0, 0, 0);
    gemm(f_v, C_, wb, C_, nullptr, t_b, C_, B_ * N_, C_, C_, 1, 0, 0, 0, 0);
    gemm(f_v, C_, wu, C_, nullptr, t_u, C_, B_ * N_, C_, C_, 1, 0, 0, 0, 0);
    gemm(f_v, C_, wv, C_, nullptr, t_v, C_, B_ * N_, C_, C_, 1, 0, 0, 0, 0);
    gemm(f_e, C_, we, C_, nullptr, t_e, C_, B_ * E_, C_, C_, 1, 0, 0, 0, 0);
    gnn_edge_kernel<<<E_, 256, 0, stream>>>(t_a, t_b, t_e, f_e);
    gnn_msg_kernel<<<B_ * N_ * (C_ / 256), 256, 0, stream>>>(f_e, t_v, msg);
    gnn_node_kernel<<<N_, 256, 0, stream>>>(t_u, msg, f_v);
  };
  gnn_layer(g_w[0], g_w[1], g_w[2], g_w[3], g_w[4]);
  gnn_layer(g_w[5], g_w[6], g_w[7], g_w[8], g_w[9]);
  // 10) outputs: cl (96) then cl_edge (8*144*4)
  classify_kernel<<<(B_ * N_ + 7) / 8, 256, 0, stream>>>(f_v, sc, out);
  gemm(f_e, C_, efc_w, 4, efc_b, out + B_ * N_, 4, B_ * E_, 4, C_, 1, 0, 0, 0, 0);
}